// ItemKNN_62405874811872
// MI455X (gfx1250) — compile-verified
//
#include <hip/hip_runtime.h>

typedef __attribute__((ext_vector_type(2))) float v2f;
typedef __attribute__((ext_vector_type(4))) float v4f;
typedef __attribute__((ext_vector_type(8))) float v8f;
typedef __attribute__((ext_vector_type(4))) int   v4i;

#define N_ITEMS_LOG2 14   // R row stride = 16384 floats
#define K_NBR 200
#define K_HALF 100

__global__ __launch_bounds__(256) void itemknn_wmma_kernel(
    const int*   __restrict__ u_arr,
    const int*   __restrict__ i_arr,
    const int*   __restrict__ nbr_idx,
    const float* __restrict__ nbr_sim,
    const float* __restrict__ R,
    float*       __restrict__ out,
    int B)
{
    const int tid   = blockIdx.x * blockDim.x + threadIdx.x;
    const int wave  = tid >> 5;                 // global wave id (wave32)
    const int lane  = threadIdx.x & 31;
    const int qbase = wave * 16;                // 16 queries per wave
    if (qbase >= B) return;

    if (qbase + 16 <= B) {
        // ---- full tile: 16 queries, 2 lanes per query (K split in halves) ----
        const int m = lane & 15;                // query within tile
        const int h = lane >> 4;                // which half of K
        const int q = qbase + m;

        const int uu = u_arr[q];
        const int ii = i_arr[q];
        const float* __restrict__ Rrow = R + ((size_t)uu << N_ITEMS_LOG2);
        const int base = ii * K_NBR + h * K_HALF;   // byte offset is 16B aligned
        const int*   __restrict__ idxp = nbr_idx + base;
        const float* __restrict__ simp = nbr_sim + base;

        float acc = 0.0f;
        #pragma unroll 5
        for (int t = 0; t < K_HALF; t += 4) {
            v4i idx4 = *(const v4i*)(idxp + t);     // 16B coalesced-per-lane load
            v4f sim4 = *(const v4f*)(simp + t);     // 16B load
            acc = fmaf(sim4.x, Rrow[idx4.x], acc);  // 4 independent R gathers
            acc = fmaf(sim4.y, Rrow[idx4.y], acc);
            acc = fmaf(sim4.z, Rrow[idx4.z], acc);
            acc = fmaf(sim4.w, Rrow[idx4.w], acc);
        }

        // ---- WMMA reduction:  D = A x Ones + 0 ----
        // A (16x4 f32): lanes 0-15 supply A[m][0] (VGPR0) / A[m][1] (VGPR1);
        //               lanes 16-31 supply A[m][2] / A[m][3].
        // Put each lane's partial in slot K0/K2, zero K1/K3. With B == all-ones:
        //   D[m][n] = A[m][0] + A[m][2]  (the exact f32 per-query sum, every column n)
        v2f a; a.x = acc;  a.y = 0.0f;
        v2f b; b.x = 1.0f; b.y = 1.0f;
        v8f c = {};
        v8f d = __builtin_amdgcn_wmma_f32_16x16x4_f32(
                    /*neg_a=*/false, a, /*neg_b=*/false, b,
                    /*c_mod=*/(short)0, c, /*reuse_a=*/false, /*reuse_b=*/false);

        // D layout: VGPR v, lanes 0-15 -> row M=v (col N=lane); lanes 16-31 -> row M=v+8.
        // All columns are identical, so pick lane==v (resp. lane==16+v) per row.
        #pragma unroll
        for (int v = 0; v < 8; ++v) {
            if (lane == v)      out[qbase + v]     = d[v];
            if (lane == 16 + v) out[qbase + 8 + v] = d[v];
        }
    } else {
        // ---- tail (not taken for B=131072): lanes 0-15 do one query each, no WMMA ----
        const int q = qbase + lane;
        if (lane < 16 && q < B) {
            const int uu = u_arr[q];
            const int ii = i_arr[q];
            const float* __restrict__ Rrow = R + ((size_t)uu << N_ITEMS_LOG2);
            const int base = ii * K_NBR;
            float acc = 0.0f;
            for (int k = 0; k < K_NBR; ++k)
                acc = fmaf(nbr_sim[base + k], Rrow[nbr_idx[base + k]], acc);
            out[q] = acc;
        }
    }
}

extern "C" void kernel_launch(void* const* d_in, const int* in_sizes, int n_in,
                              void* d_out, int out_size, void* d_ws, size_t ws_size,
                              hipStream_t stream)
{
    const int*   u  = (const int*)  d_in[0];
    const int*   it = (const int*)  d_in[1];
    const int*   ni = (const int*)  d_in[2];   // nbr_idx (N_ITEMS x 200)
    const float* ns = (const float*)d_in[3];   // nbr_sim (N_ITEMS x 200)
    const float* R  = (const float*)d_in[4];   // R (N_USERS x N_ITEMS)
    float* out = (float*)d_out;

    const int B = in_sizes[0];
    // 256 threads = 8 waves = 128 queries per block
    const int queriesPerBlock = 128;
    const int blocks = (B + queriesPerBlock - 1) / queriesPerBlock;
    itemknn_wmma_kernel<<<blocks, 256, 0, stream>>>(u, it, ni, ns, R, out, B);
}